// Post_rois_28381143892896
// MI455X (gfx1250) — compile-verified
//
#include <hip/hip_runtime.h>
#include <cstdint>
#include <cstddef>

#define NCLS      2
#define TOPK      200
#define CONF_T    0.01f
#define NMS_T     0.45f
#define NPRIORS   4096
#define NBATCH    8

// ---------------------------------------------------------------------------
// CDNA5 async global->LDS DMA path (ASYNCcnt).  The round-1 diagnostic showed
// the builtins take (v2i/v4i addrspace(1)*, v2i/v4i addrspace(3)*, Ii, Ii).
// ---------------------------------------------------------------------------
#if defined(__gfx1250__) &&                                             \
    __has_builtin(__builtin_amdgcn_global_load_async_to_lds_b64) &&     \
    __has_builtin(__builtin_amdgcn_global_load_async_to_lds_b128) &&    \
    __has_builtin(__builtin_amdgcn_s_wait_asynccnt)
#define USE_ASYNC 1
#else
#define USE_ASYNC 0
#endif

#if USE_ASYNC
typedef int v2i_t __attribute__((__vector_size__(2 * sizeof(int))));
typedef int v4i_t __attribute__((__vector_size__(4 * sizeof(int))));
typedef __attribute__((address_space(1))) v2i_t* g2p;   // global (__device__)
typedef __attribute__((address_space(3))) v2i_t* l2p;   // LDS (__shared__)
typedef __attribute__((address_space(1))) v4i_t* g4p;
typedef __attribute__((address_space(3))) v4i_t* l4p;
#endif

// ---------------------------------------------------------------------------
// Kernel 1: softmax score + box decode + sort-key pack; also zero the output.
//   key = (~bits(score) << 32) | prior_idx   -> ascending u64 sort gives
//   score-descending, index-ascending (matches stable argsort(-score)).
// ---------------------------------------------------------------------------
__global__ void decode_kernel(const float* __restrict__ loc,
                              const float* __restrict__ conf,
                              const float* __restrict__ prior,
                              unsigned long long* __restrict__ keysW,
                              float4* __restrict__ boxW,
                              float* __restrict__ out, int out_n) {
  const int i = blockIdx.x * blockDim.x + threadIdx.x;   // 0 .. B*N-1
  if (i < out_n) out[i] = 0.0f;                           // harness poisons out
  if (i >= NBATCH * NPRIORS) return;

  // gfx1250 prefetch (global_prefetch_b8) for the next block's loc tile.
  __builtin_prefetch(loc + 4 * (i + blockDim.x), 0, 1);

  const int p = i & (NPRIORS - 1);

  const float c0 = conf[2 * i + 0];
  const float c1 = conf[2 * i + 1];
  const float score = 1.0f / (1.0f + expf(c0 - c1));      // softmax class-1

  const float4 pr = ((const float4*)prior)[p];            // px,py,pw,ph
  const float4 l  = ((const float4*)loc)[i];
  const float cx = pr.x + l.x * 0.1f * pr.z;
  const float cy = pr.y + l.y * 0.1f * pr.w;
  const float w  = pr.z * expf(l.z * 0.2f);
  const float h  = pr.w * expf(l.w * 0.2f);
  boxW[i] = make_float4(cx - 0.5f * w, cy - 0.5f * h,
                        cx + 0.5f * w, cy + 0.5f * h);

  const unsigned sb = __float_as_uint(score);
  keysW[i] = ((unsigned long long)(~sb) << 32) | (unsigned)p;
}

// ---------------------------------------------------------------------------
// Kernel 2: one workgroup (32 wave32s) per image.
//   LDS: keys 32KB | sorted boxes 64KB | suppression bitmap 512B  (~97KB of
//   the 320KB/WGP LDS).  Bitonic sort, async gather, lockstep greedy NMS.
// ---------------------------------------------------------------------------
__global__ void __launch_bounds__(1024, 1)
nms_kernel(const unsigned long long* __restrict__ keysW,
           const float4* __restrict__ boxW,
           float* __restrict__ out) {
  const int img = blockIdx.x;
  const int tid = threadIdx.x;
  const int NT  = blockDim.x;

  extern __shared__ char smem[];
  unsigned long long* keys = (unsigned long long*)smem;                      // 32768 B
  float4*             sbox = (float4*)(smem + NPRIORS * 8);                  // 65536 B
  unsigned*           supp = (unsigned*)(smem + NPRIORS * 8 + NPRIORS * 16); //   512 B

  // ---- stage keys into LDS (async DMA, ASYNCcnt) --------------------------
#if USE_ASYNC
  for (int r = tid; r < NPRIORS; r += NT)
    __builtin_amdgcn_global_load_async_to_lds_b64(
        (g2p)(keysW + (size_t)img * NPRIORS + r), (l2p)(keys + r), 0, 0);
  __builtin_amdgcn_s_wait_asynccnt(0);
#else
  for (int r = tid; r < NPRIORS; r += NT)
    keys[r] = keysW[(size_t)img * NPRIORS + r];
#endif
  for (int w = tid; w < NPRIORS / 32; w += NT) supp[w] = 0u;
  __syncthreads();

  // ---- bitonic sort, ascending on transformed key -------------------------
  for (int k = 2; k <= NPRIORS; k <<= 1) {
    for (int j = k >> 1; j > 0; j >>= 1) {
      for (int i = tid; i < NPRIORS; i += NT) {
        const int ixj = i ^ j;
        if (ixj > i) {
          const bool up = ((i & k) == 0);
          const unsigned long long a = keys[i], b = keys[ixj];
          if ((a > b) == up) { keys[i] = b; keys[ixj] = a; }
        }
      }
      __syncthreads();
    }
  }

  // ---- gather boxes into sorted order (indirect async DMA, b128) ----------
#if USE_ASYNC
  for (int r = tid; r < NPRIORS; r += NT) {
    const int src = (int)(keys[r] & 0xFFFFFFFFull);
    __builtin_amdgcn_global_load_async_to_lds_b128(
        (g4p)(boxW + (size_t)img * NPRIORS + src), (l4p)(sbox + r), 0, 0);
  }
  __builtin_amdgcn_s_wait_asynccnt(0);
#else
  for (int r = tid; r < NPRIORS; r += NT) {
    const int src = (int)(keys[r] & 0xFFFFFFFFull);
    sbox[r] = boxW[(size_t)img * NPRIORS + src];
  }
#endif
  __syncthreads();

  // ---- lockstep greedy NMS ------------------------------------------------
  // All control state (r, kept) is recomputed identically by every thread
  // from LDS that is stable at each __syncthreads boundary.
  int kept = 0;
  int r = 0;
  while (kept < TOPK) {
    while (r < NPRIORS && ((supp[r >> 5] >> (r & 31)) & 1u)) ++r;
    if (r >= NPRIORS) break;

    const unsigned long long krec = keys[r];
    const float score = __uint_as_float(~(unsigned)(krec >> 32));
    if (!(score > CONF_T)) break;   // sorted descending -> nothing valid left

    const float4 kb = sbox[r];
    if (tid == 0) {
      float* o = out + ((((size_t)img * NCLS) + 1) * TOPK + kept) * 5;
      o[0] = score; o[1] = kb.x; o[2] = kb.y; o[3] = kb.z; o[4] = kb.w;
    }

    const float karea = (kb.z - kb.x) * (kb.w - kb.y);
    for (int j = r + 1 + tid; j < NPRIORS; j += NT) {
      if (((supp[j >> 5] >> (j & 31)) & 1u) == 0u) {
        const float4 b = sbox[j];
        float iw = fminf(kb.z, b.z) - fmaxf(kb.x, b.x);
        float ih = fminf(kb.w, b.w) - fmaxf(kb.y, b.y);
        iw = fmaxf(iw, 0.0f);
        ih = fmaxf(ih, 0.0f);
        const float inter = iw * ih;
        const float barea = (b.z - b.x) * (b.w - b.y);
        const float iou = inter / (karea + barea - inter);
        if (iou > NMS_T) atomicOr(&supp[j >> 5], 1u << (j & 31));
      }
    }
    ++kept;
    ++r;
    __syncthreads();
  }
}

// ---------------------------------------------------------------------------
extern "C" void kernel_launch(void* const* d_in, const int* in_sizes, int n_in,
                              void* d_out, int out_size, void* d_ws, size_t ws_size,
                              hipStream_t stream) {
  const float* loc   = (const float*)d_in[0];   // (8, 4096, 4) f32
  const float* conf  = (const float*)d_in[1];   // (32768, 2)   f32
  const float* prior = (const float*)d_in[2];   // (4096, 4)    f32
  float* out = (float*)d_out;                   // (8, 2, 200, 5) f32

  unsigned long long* keysW = (unsigned long long*)d_ws;                // 256 KB
  float4* boxW = (float4*)((char*)d_ws + (size_t)NBATCH * NPRIORS * 8); // 512 KB

  const int total = NBATCH * NPRIORS;
  decode_kernel<<<(total + 255) / 256, 256, 0, stream>>>(
      loc, conf, prior, keysW, boxW, out, out_size);

  const size_t shmem = (size_t)NPRIORS * 8 + (size_t)NPRIORS * 16 + (NPRIORS / 32) * 4;
  // >64KB dynamic LDS: opt in (deterministic, not a stream op, capture-safe).
  (void)hipFuncSetAttribute((const void*)nms_kernel,
                            hipFuncAttributeMaxDynamicSharedMemorySize,
                            (int)shmem);
  nms_kernel<<<NBATCH, 1024, shmem, stream>>>(keysW, boxW, out);
}